// GATv2Layer_35648228557643
// MI455X (gfx1250) — compile-verified
//
#include <hip/hip_runtime.h>

typedef __attribute__((ext_vector_type(2))) float v2f;
typedef __attribute__((ext_vector_type(8))) float v8f;

#define NEG_SLOPE 0.2f
#define BN_EPS 1e-5f

__device__ __forceinline__ float lrelu(float x) { return x > 0.0f ? x : NEG_SLOPE * x; }

// order-preserving float <-> uint encoding so atomicMax(uint) == float max.
// init value 0x00000000 decodes below every real float (it maps to -NaN).
__device__ __forceinline__ unsigned fenc(float f) {
    unsigned u = __float_as_uint(f);
    return (u & 0x80000000u) ? ~u : (u | 0x80000000u);
}
__device__ __forceinline__ float fdec(unsigned u) {
    return __uint_as_float((u & 0x80000000u) ? (u ^ 0x80000000u) : ~u);
}

// ---------------------------------------------------------------- zero scratch
__global__ void zero_ws(float* __restrict__ p, int n) {
    int i = blockIdx.x * blockDim.x + threadIdx.x;
    if (i < n) p[i] = 0.0f;
}

// ------------------------------------------------- xl = x@Wl, xr = x@Wr (WMMA)
// grid = N/16 blocks of 256 (8 waves). Wave w owns output cols [16w,16w+16).
// A 16x4 f32 layout: lanes 0..15 = M, VGPR0/1 = K{0,1} (low half) K{2,3} (high)
// B 4x16  f32 layout: lanes 0..15 = N, VGPR0/1 = K{0,1} / K{2,3}
// C 16x16 f32 layout: VGPR r lane l -> (M = r + (l>=16 ? 8:0), N = l&15)
__global__ __launch_bounds__(256) void node_transform(
    const float* __restrict__ x, const float* __restrict__ Wl,
    const float* __restrict__ Wr, float* __restrict__ xl, float* __restrict__ xr,
    int N) {
    const int wave = threadIdx.x >> 5;
    const int lane = threadIdx.x & 31;
    const int lo   = lane & 15;
    const int hiK  = (lane >= 16) ? 2 : 0;
    const int hiM  = (lane >= 16) ? 8 : 0;
    const int rowTile = blockIdx.x * 16;
    const int col  = wave * 16 + lo;
    const int aRow = rowTile + lo;

    v8f cl = {};
    v8f cr = {};
    for (int k = 0; k < 128; k += 4) {
        v2f a = *(const v2f*)&x[aRow * 128 + k + hiK];
        v2f bl, br;
        bl.x = Wl[(k + hiK) * 128 + col];
        bl.y = Wl[(k + hiK + 1) * 128 + col];
        br.x = Wr[(k + hiK) * 128 + col];
        br.y = Wr[(k + hiK + 1) * 128 + col];
        cl = __builtin_amdgcn_wmma_f32_16x16x4_f32(false, a, false, bl, (short)0, cl, false, false);
        cr = __builtin_amdgcn_wmma_f32_16x16x4_f32(false, a, false, br, (short)0, cr, false, false);
    }
#pragma unroll
    for (int r = 0; r < 8; ++r) {
        int row = rowTile + r + hiM;
        xl[row * 128 + col] = cl[r];
        xr[row * 128 + col] = cr[r];
    }
}

// ------------------------------------- per-edge attention logits (WMMA, fused)
// One wave handles 16 edges: C init = xl[src]+xr[dst] gathers, 4 WMMA K-steps
// accumulate edge_attr@We, then lrelu * att reduced over cols per edge.
// grid = E/128 blocks of 256 (8 waves, each its own 16-edge tile).
__global__ __launch_bounds__(256) void edge_logits(
    const float* __restrict__ edge_attr, const float* __restrict__ We,
    const float* __restrict__ xl, const float* __restrict__ xr,
    const int* __restrict__ srcIdx, const int* __restrict__ dstIdx,
    const float* __restrict__ att, float* __restrict__ ebuf,
    unsigned* __restrict__ menc, int E) {
    __shared__ float sWe[16 * 128];
    for (int i = threadIdx.x; i < 16 * 128; i += 256) sWe[i] = We[i];
    __syncthreads();

    const int wave = threadIdx.x >> 5;
    const int lane = threadIdx.x & 31;
    const int lo   = lane & 15;
    const int hiK  = (lane >= 16) ? 2 : 0;
    const int hiM  = (lane >= 16) ? 8 : 0;
    const int tile = (blockIdx.x * 8 + wave) * 16;  // first edge of this wave

    // A fragments for K = 0..15 (edge_attr rows are 16 floats)
    const float* aRow = &edge_attr[(size_t)(tile + lo) * 16];
    v2f a0 = *(const v2f*)&aRow[0 + hiK];
    v2f a1 = *(const v2f*)&aRow[4 + hiK];
    v2f a2 = *(const v2f*)&aRow[8 + hiK];
    v2f a3 = *(const v2f*)&aRow[12 + hiK];

    // lane lo holds indices of edge tile+lo; broadcast so every lane knows the
    // 8 edges of its half (rows r + hiM)
    int sIdx = srcIdx[tile + lo];
    int dIdx = dstIdx[tile + lo];
    int myS[8], myD[8];
#pragma unroll
    for (int r = 0; r < 8; ++r) {
        myS[r] = __shfl(sIdx, r + hiM, 32);
        myD[r] = __shfl(dIdx, r + hiM, 32);
    }

    float acc[8];
#pragma unroll
    for (int r = 0; r < 8; ++r) acc[r] = 0.0f;

#pragma unroll
    for (int wi = 0; wi < 8; ++wi) {
        const int col = wi * 16 + lo;
        v8f c;
#pragma unroll
        for (int r = 0; r < 8; ++r)
            c[r] = xl[(size_t)myS[r] * 128 + col] + xr[(size_t)myD[r] * 128 + col];
        v2f b;
        b.x = sWe[(0 + hiK) * 128 + col];  b.y = sWe[(1 + hiK) * 128 + col];
        c = __builtin_amdgcn_wmma_f32_16x16x4_f32(false, a0, false, b, (short)0, c, false, false);
        b.x = sWe[(4 + hiK) * 128 + col];  b.y = sWe[(5 + hiK) * 128 + col];
        c = __builtin_amdgcn_wmma_f32_16x16x4_f32(false, a1, false, b, (short)0, c, false, false);
        b.x = sWe[(8 + hiK) * 128 + col];  b.y = sWe[(9 + hiK) * 128 + col];
        c = __builtin_amdgcn_wmma_f32_16x16x4_f32(false, a2, false, b, (short)0, c, false, false);
        b.x = sWe[(12 + hiK) * 128 + col]; b.y = sWe[(13 + hiK) * 128 + col];
        c = __builtin_amdgcn_wmma_f32_16x16x4_f32(false, a3, false, b, (short)0, c, false, false);

        const float av = att[col];
#pragma unroll
        for (int r = 0; r < 8; ++r) acc[r] += lrelu(c[r]) * av;
    }

    // reduce over the 16 lanes of each half (xor masks stay inside the half)
#pragma unroll
    for (int r = 0; r < 8; ++r) {
        float v = acc[r];
        v += __shfl_xor(v, 1, 32);
        v += __shfl_xor(v, 2, 32);
        v += __shfl_xor(v, 4, 32);
        v += __shfl_xor(v, 8, 32);
        if (lo == 0) {
            int edge = tile + r + hiM;
            ebuf[edge] = v;
            atomicMax(&menc[myD[r]], fenc(v));
        }
    }
}

// -------------------------------------------------- p = exp(e - m[dst]); sums
__global__ void edge_exp(const float* __restrict__ ebuf,
                         const int* __restrict__ dstIdx,
                         const unsigned* __restrict__ menc,
                         float* __restrict__ pbuf, float* __restrict__ denom,
                         int E) {
    int i = blockIdx.x * blockDim.x + threadIdx.x;
    if (i >= E) return;
    int d = dstIdx[i];
    float p = expf(ebuf[i] - fdec(menc[d]));
    pbuf[i] = p;
    atomicAdd(&denom[d], p);
}

// -------------------------------------- out[dst] += alpha * xl[src] (scatter)
__global__ __launch_bounds__(256) void edge_scatter(
    const float* __restrict__ xl, const float* __restrict__ pbuf,
    const float* __restrict__ denom, const int* __restrict__ srcIdx,
    const int* __restrict__ dstIdx, float* __restrict__ accum, int E) {
    int gw   = (blockIdx.x * 256 + threadIdx.x) >> 5;  // one wave per edge
    int lane = threadIdx.x & 31;
    if (gw >= E) return;
    int d = dstIdx[gw];
    float alpha = pbuf[gw] / denom[d];
    int s = srcIdx[gw];
    const float4 v = *(const float4*)&xl[(size_t)s * 128 + lane * 4];
    float* o = &accum[(size_t)d * 128 + lane * 4];
    atomicAdd(o + 0, alpha * v.x);
    atomicAdd(o + 1, alpha * v.y);
    atomicAdd(o + 2, alpha * v.z);
    atomicAdd(o + 3, alpha * v.w);
}

// ------------------------- bias + lrelu, write out, per-channel sum / sumsq
__global__ __launch_bounds__(128) void bias_act_stats(
    const float* __restrict__ accum, const float* __restrict__ bias,
    float* __restrict__ out, float* __restrict__ stats, int N, int rowsPerBlock) {
    int c  = threadIdx.x;
    int r0 = blockIdx.x * rowsPerBlock;
    int r1 = min(r0 + rowsPerBlock, N);
    float b = bias[c];
    float sum = 0.0f, sq = 0.0f;
    for (int r = r0; r < r1; ++r) {
        float v = lrelu(accum[(size_t)r * 128 + c] + b);
        out[(size_t)r * 128 + c] = v;
        sum += v;
        sq  += v * v;
    }
    atomicAdd(&stats[c], sum);
    atomicAdd(&stats[128 + c], sq);
}

// ---------------------------------------------------- batchnorm scale / shift
__global__ void bn_prep(const float* __restrict__ stats,
                        const float* __restrict__ bnw,
                        const float* __restrict__ bnb,
                        float* __restrict__ scsh, int N) {
    int c = threadIdx.x;
    float inv = 1.0f / (float)N;
    float mean = stats[c] * inv;
    float var  = stats[128 + c] * inv - mean * mean;
    float sc   = bnw[c] * rsqrtf(var + BN_EPS);
    scsh[c]       = sc;
    scsh[128 + c] = bnb[c] - mean * sc;
}

__global__ void bn_apply(float* __restrict__ out, const float* __restrict__ scsh,
                         int total) {
    int i = blockIdx.x * blockDim.x + threadIdx.x;
    if (i >= total) return;
    int c = i & 127;
    out[i] = out[i] * scsh[c] + scsh[128 + c];
}

extern "C" void kernel_launch(void* const* d_in, const int* in_sizes, int n_in,
                              void* d_out, int out_size, void* d_ws, size_t ws_size,
                              hipStream_t stream) {
    const float* x    = (const float*)d_in[0];
    const float* ea   = (const float*)d_in[1];
    const int*   ei   = (const int*)d_in[2];
    const float* Wl   = (const float*)d_in[3];
    const float* Wr   = (const float*)d_in[4];
    const float* We   = (const float*)d_in[5];
    const float* att  = (const float*)d_in[6];
    const float* bias = (const float*)d_in[7];
    const float* bnw  = (const float*)d_in[8];
    const float* bnb  = (const float*)d_in[9];
    float* out = (float*)d_out;

    const int N = in_sizes[0] / 128;  // 10000 (multiple of 16)
    const int E = in_sizes[2] / 2;    // 640000 (multiple of 128)
    const int* srcIdx = ei;
    const int* dstIdx = ei + E;

    // workspace layout
    float* ws = (float*)d_ws;
    float* xl = ws;     ws += (size_t)N * 128;
    float* xr = ws;     ws += (size_t)N * 128;
    float* ebuf = ws;   ws += E;
    float* pbuf = ws;   ws += E;
    float* zbase = ws;  // contiguous zero-init region starts here
    unsigned* menc = (unsigned*)ws; ws += N;
    float* denom = ws;  ws += N;
    float* accum = ws;  ws += (size_t)N * 128;
    float* stats = ws;  ws += 256;
    float* scsh  = ws;  ws += 256;
    int zcount = (int)(ws - zbase);

    zero_ws<<<(zcount + 255) / 256, 256, 0, stream>>>(zbase, zcount);
    node_transform<<<N / 16, 256, 0, stream>>>(x, Wl, Wr, xl, xr, N);
    edge_logits<<<E / 128, 256, 0, stream>>>(ea, We, xl, xr, srcIdx, dstIdx, att,
                                             ebuf, menc, E);
    edge_exp<<<(E + 255) / 256, 256, 0, stream>>>(ebuf, dstIdx, menc, pbuf, denom, E);
    edge_scatter<<<(E + 7) / 8, 256, 0, stream>>>(xl, pbuf, denom, srcIdx, dstIdx,
                                                  accum, E);
    const int rpb = 50;
    bias_act_stats<<<(N + rpb - 1) / rpb, 128, 0, stream>>>(accum, bias, out, stats,
                                                            N, rpb);
    bn_prep<<<1, 128, 0, stream>>>(stats, bnw, bnb, scsh, N);
    bn_apply<<<(N * 128 + 255) / 256, 256, 0, stream>>>(out, scsh, N * 128);
}